// Model_68384469287338
// MI455X (gfx1250) — compile-verified
//
#include <hip/hip_runtime.h>
#include <hip/hip_bf16.h>
#include <math.h>

// ---------------------------------------------------------------------------
// Model constants
// ---------------------------------------------------------------------------
#define BATCH    32
#define SEQ_LEN  336
#define PRED_LEN 7
#define D_MODEL  512
#define D_STATE  256
#define D_FF     2048
#define N_VARS   32
#define V_EXO    31          // N_VARS - 1
#define P_PATCH  48
#define NTOK     49          // 1 + P_PATCH
#define NROWS    (BATCH * NTOK)       // 1568 token rows
#define NSEQ     (BATCH * V_EXO)      // 992 exo sequences
#define D_EXO    64
#define DS_EXO   128
#define DTRK_E   4           // exo dt_rank
#define DTRK     32          // encoder dt_rank
#define DBC_E    260         // 4 + 128 + 128
#define DBC_M    544         // 32 + 256 + 256
#define LCH      84          // exo scan L-chunk (336 / 4)

typedef float v2f __attribute__((ext_vector_type(2)));
typedef float v8f __attribute__((ext_vector_type(8)));

__device__ __forceinline__ float siluf(float x) { return x / (1.f + __expf(-x)); }
__device__ __forceinline__ float softplusf(float x) { return x > 20.f ? x : log1pf(__expf(x)); }

// ---------------------------------------------------------------------------
// Generic fp32 WMMA GEMM:  C[M,N] = act(A[M,K] @ B + bias)
// B is K-major (ldb=N-ish) when transB==0, or N-major (B[n*ldb+k]) when transB==1.
// Tile: 64x64 per workgroup, 8 waves; each wave owns 16 rows x 32 cols
// (two v_wmma_f32_16x16x4_f32 accumulators), K staged 32-deep through LDS.
// act: 0=none, 1=relu, 2=softplus
// ---------------------------------------------------------------------------
#define GEMM_KC 32
__global__ __launch_bounds__(256) void gemm_wmma_f32(
    const float* __restrict__ A, int lda,
    const float* __restrict__ B, int ldb, int transB,
    const float* __restrict__ bias,
    float* __restrict__ C, int ldc,
    int M, int N, int K, int act)
{
  __shared__ float As[GEMM_KC][64 + 1];   // As[k][m]
  __shared__ float Bs[GEMM_KC][64 + 1];   // Bs[k][n]
  const int tid  = threadIdx.x;
  const int lane = tid & 31;
  const int wave = tid >> 5;
  const int m0 = (int)blockIdx.x * 64;
  const int n0 = (int)blockIdx.y * 64;
  const int wm = (wave >> 1) * 16;        // 0,16,32,48
  const int wn = (wave & 1) * 32;         // 0,32
  const int half = lane >> 4;             // 0/1 -> K lanes 0,1 vs 2,3
  const int l15  = lane & 15;

  v8f acc0 = {}; v8f acc1 = {};

  for (int k0 = 0; k0 < K; k0 += GEMM_KC) {
    // stage A (coalesced on K)
    for (int i = tid; i < GEMM_KC * 64; i += 256) {
      int kk = i & 31, mm = i >> 5;
      int gm = m0 + mm, gk = k0 + kk;
      As[kk][mm] = (gm < M && gk < K) ? A[(size_t)gm * lda + gk] : 0.f;
    }
    // stage B
    if (!transB) {
      for (int i = tid; i < GEMM_KC * 64; i += 256) {
        int nn = i & 63, kk = i >> 6;
        int gn = n0 + nn, gk = k0 + kk;
        Bs[kk][nn] = (gn < N && gk < K) ? B[(size_t)gk * ldb + gn] : 0.f;
      }
    } else {
      for (int i = tid; i < GEMM_KC * 64; i += 256) {
        int kk = i & 31, nn = i >> 5;
        int gn = n0 + nn, gk = k0 + kk;
        Bs[kk][nn] = (gn < N && gk < K) ? B[(size_t)gn * ldb + gk] : 0.f;
      }
    }
    __syncthreads();
#pragma unroll
    for (int kk = 0; kk < GEMM_KC; kk += 4) {
      const int kr = kk + (half << 1);
      v2f a, b0, b1;
      a.x  = As[kr][wm + l15];      a.y  = As[kr + 1][wm + l15];
      b0.x = Bs[kr][wn + l15];      b0.y = Bs[kr + 1][wn + l15];
      b1.x = Bs[kr][wn + 16 + l15]; b1.y = Bs[kr + 1][wn + 16 + l15];
      acc0 = __builtin_amdgcn_wmma_f32_16x16x4_f32(false, a, false, b0, (short)0, acc0, false, false);
      acc1 = __builtin_amdgcn_wmma_f32_16x16x4_f32(false, a, false, b1, (short)0, acc1, false, false);
    }
    __syncthreads();
  }
  // C layout: VGPR r -> row (wm + r + 8*half), col (wn + l15) / (+16)
#pragma unroll
  for (int r = 0; r < 8; ++r) {
    int row = m0 + wm + r + (half << 3);
    if (row >= M) continue;
    int c0 = n0 + wn + l15;
    int c1 = c0 + 16;
    float v0 = acc0[r], v1 = acc1[r];
    if (bias) {
      if (c0 < N) v0 += bias[c0];
      if (c1 < N) v1 += bias[c1];
    }
    if (act == 1)      { v0 = fmaxf(v0, 0.f); v1 = fmaxf(v1, 0.f); }
    else if (act == 2) { v0 = softplusf(v0);  v1 = softplusf(v1);  }
    if (c0 < N) C[(size_t)row * ldc + c0] = v0;
    if (c1 < N) C[(size_t)row * ldc + c1] = v1;
  }
}

// ---------------------------------------------------------------------------
// Prep: rank-1 collapse of exo input proj (u2/c2), fused out-proj W2, query q
// ---------------------------------------------------------------------------
__global__ __launch_bounds__(256) void prep_kernel(
    const float* __restrict__ exo_in_w,   // (1,64)
    const float* __restrict__ exo_in_b,   // (64)
    const float* __restrict__ exo_m_in_w, // (64,128)
    const float* __restrict__ m_out_w,    // (64,64)
    const float* __restrict__ e_out_w,    // (64,512)
    const float* __restrict__ g_endo,     // (512)
    const float* __restrict__ fuse_in_w,  // (1536,512)
    const float* __restrict__ fuse_in_b,  // (1536)
    float* __restrict__ u2c2,             // [256]: u2[128] | c2[128]
    float* __restrict__ W2,               // [64*512]
    float* __restrict__ qv)               // [512]
{
  int tid = threadIdx.x;
  for (int j = tid; j < 128; j += 256) {
    float a = 0.f, c = 0.f;
    for (int d = 0; d < 64; ++d) {
      float w = exo_m_in_w[d * 128 + j];
      a += exo_in_w[d] * w;
      c += exo_in_b[d] * w;
    }
    u2c2[j] = a; u2c2[128 + j] = c;
  }
  for (int o = tid; o < 512; o += 256) {
    float a = fuse_in_b[o];                 // bq
    for (int d = 0; d < 512; ++d) a += g_endo[d] * fuse_in_w[(size_t)o * 512 + d];
    qv[o] = a;
  }
  for (int i = tid; i < 64 * 512; i += 256) {
    int d = i >> 9, o = i & 511;
    float a = 0.f;
    for (int k = 0; k < 64; ++k) a += m_out_w[d * 64 + k] * e_out_w[(size_t)k * 512 + o];
    W2[i] = a;
  }
}

// ---------------------------------------------------------------------------
// Endo patch embedding -> token rows b*49 + 1 + p
// ---------------------------------------------------------------------------
__global__ __launch_bounds__(256) void endo_tokens_kernel(
    const float* __restrict__ x_enc, const float* __restrict__ proj_w,
    const float* __restrict__ proj_b, const float* __restrict__ pos,
    float* __restrict__ tokens)
{
  size_t idx = (size_t)blockIdx.x * 256 + threadIdx.x;
  if (idx >= (size_t)BATCH * P_PATCH * D_MODEL) return;
  int d = idx & 511;
  int bp = (int)(idx >> 9);
  int b = bp / P_PATCH, p = bp % P_PATCH;
  float acc = proj_b[d] + pos[p * D_MODEL + d];
#pragma unroll
  for (int k = 0; k < 7; ++k) {
    float v = x_enc[((size_t)b * SEQ_LEN + p * 7 + k) * N_VARS];
    if (v == -9999.0f) v = 0.f;
    acc += v * proj_w[k * D_MODEL + d];
  }
  tokens[((size_t)(b * NTOK + 1 + p)) * D_MODEL + d] = acc;
}

// ---------------------------------------------------------------------------
// Exo Mamba: one workgroup per sequence; chunked LDS-resident selective scan.
// Thread map: dl = tid>>2 (channel), squad = tid&3 owns 32 states.
// Epilogue fuses z-gate, mean over L, and @W2 projection -> exo_tok row.
// ---------------------------------------------------------------------------
__global__ __launch_bounds__(256) void exo_mamba_kernel(
    const float* __restrict__ x_enc,
    const float* __restrict__ u2c2,
    const float* __restrict__ conv_w, const float* __restrict__ conv_b,
    const float* __restrict__ x_w,                       // (64,260)
    const float* __restrict__ dt_w, const float* __restrict__ dt_b,
    const float* __restrict__ A_log, const float* __restrict__ Dv,
    const float* __restrict__ W2, const float* __restrict__ out_b,
    float* __restrict__ exo_tok)
{
  extern __shared__ float sm[];
  float* ss   = sm;                  // 336
  float* xcC  = ss + SEQ_LEN;        // 84*64   = 5376
  float* dtrC = xcC + LCH * 64;      // 84*4    = 336
  float* dtC  = dtrC + LCH * 4;      // 84*64   = 5376
  float* bC   = dtC + LCH * 64;      // 84*128  = 10752
  float* cC   = bC + LCH * 128;      // 84*128  = 10752

  const int tid = threadIdx.x;
  const int p = blockIdx.x;
  const int b = p / V_EXO, v = p % V_EXO;

  for (int l = tid; l < SEQ_LEN; l += 256) {
    float val = x_enc[((size_t)b * SEQ_LEN + l) * N_VARS + 1 + v];
    ss[l] = (val == -9999.0f) ? 0.f : val;
  }

  const int dl = tid >> 2, squad = tid & 3, s0 = squad * 32;
  float h[32], Areg[32];
#pragma unroll
  for (int j = 0; j < 32; ++j) {
    h[j] = 0.f;
    Areg[j] = -__expf(A_log[dl * DS_EXO + s0 + j]);
  }
  const float Dval = Dv[dl];
  const float u2z = u2c2[64 + dl], c2z = u2c2[128 + 64 + dl];
  float msum = 0.f;
  __syncthreads();

  for (int ch = 0; ch < SEQ_LEN / LCH; ++ch) {
    const int Lb = ch * LCH;
    // xc chunk (conv + silu), from rank-1 input projection
    for (int i = tid; i < LCH * 64; i += 256) {
      int lc = i >> 6, d = i & 63;
      int l = Lb + lc;
      float u2d = u2c2[d], c2d = u2c2[128 + d];
      float xi   = ss[l] * u2d + c2d;
      float xim1 = (l > 0) ? (ss[l - 1] * u2d + c2d) : 0.f;
      float pre = conv_w[d] * xim1 + conv_w[64 + d] * xi + conv_b[d];
      xcC[i] = siluf(pre);
    }
    __syncthreads();
    // dtr = xc @ x_w[:, :4]
    for (int i = tid; i < LCH * 4; i += 256) {
      int lc = i >> 2, r = i & 3;
      float acc = 0.f;
      for (int k = 0; k < 64; ++k) acc += xcC[lc * 64 + k] * x_w[k * DBC_E + r];
      dtrC[i] = acc;
    }
    __syncthreads();
    // dt = softplus(dtr @ dt_w + dt_b)
    for (int i = tid; i < LCH * 64; i += 256) {
      int lc = i >> 6, d = i & 63;
      float acc = dt_b[d];
#pragma unroll
      for (int r = 0; r < 4; ++r) acc += dtrC[lc * 4 + r] * dt_w[r * 64 + d];
      dtC[i] = softplusf(acc);
    }
    // B, C chunks
    for (int i = tid; i < LCH * 128; i += 256) {
      int lc = i >> 7, s = i & 127;
      float aB = 0.f, aC = 0.f;
      for (int k = 0; k < 64; ++k) {
        float xv = xcC[lc * 64 + k];
        aB += xv * x_w[k * DBC_E + 4 + s];
        aC += xv * x_w[k * DBC_E + 4 + DS_EXO + s];
      }
      bC[i] = aB; cC[i] = aC;
    }
    __syncthreads();
    // selective scan over chunk
    for (int lc = 0; lc < LCH; ++lc) {
      const int l = Lb + lc;
      const float dtv = dtC[lc * 64 + dl];
      const float xv  = xcC[lc * 64 + dl];
      const float dtx = dtv * xv;
      float acc = 0.f;
      const float* bl = bC + lc * 128 + s0;
      const float* cl = cC + lc * 128 + s0;
#pragma unroll
      for (int j = 0; j < 32; ++j) {
        float dA = __expf(dtv * Areg[j]);
        h[j] = dA * h[j] + dtx * bl[j];
        acc += h[j] * cl[j];
      }
      acc += __shfl_xor(acc, 1, 32);
      acc += __shfl_xor(acc, 2, 32);
      if (squad == 0) {
        float y = acc + xv * Dval;
        float zr = ss[l] * u2z + c2z;
        msum += y * siluf(zr);
      }
    }
    __syncthreads();
  }
  // mean over L, then @ W2 (+ out_b) -> exo token
  if (squad == 0) xcC[dl] = msum * (1.f / (float)SEQ_LEN);
  __syncthreads();
  for (int o = tid; o < D_MODEL; o += 256) {
    float acc = out_b[o];
#pragma unroll 8
    for (int d = 0; d < 64; ++d) acc += xcC[d] * W2[d * D_MODEL + o];
    exo_tok[(size_t)p * D_MODEL + o] = acc;
  }
}

// ---------------------------------------------------------------------------
// Fuse: cross attention (1 query, 31 kv, 8 heads) + out-proj + residual + LN
// One workgroup per batch -> writes tokens row b*49.
// ---------------------------------------------------------------------------
__global__ __launch_bounds__(256) void fuse_attn_kernel(
    const float* __restrict__ qv, const float* __restrict__ kbuf,
    const float* __restrict__ vbuf, const float* __restrict__ out_w,
    const float* __restrict__ out_b, const float* __restrict__ g_endo,
    const float* __restrict__ ln_g, const float* __restrict__ ln_b,
    float* __restrict__ tokens)
{
  __shared__ float sc[8 * 32];
  __shared__ float att[8 * 32];
  __shared__ float ov[512];
  __shared__ float rbuf[512];
  __shared__ float rs[8], rs2[8], mv[2];
  const int b = blockIdx.x, tid = threadIdx.x;

  if (tid < 8 * V_EXO) {
    int hh = tid / V_EXO, i = tid % V_EXO;
    const float* kk = kbuf + ((size_t)(b * V_EXO + i)) * 512 + hh * 64;
    const float* qq = qv + hh * 64;
    float a = 0.f;
#pragma unroll 8
    for (int j = 0; j < 64; ++j) a += qq[j] * kk[j];
    sc[hh * 32 + i] = a * 0.125f;   // 1/sqrt(64)
  }
  __syncthreads();
  if (tid < 8) {
    float m = -1e30f;
    for (int i = 0; i < V_EXO; ++i) m = fmaxf(m, sc[tid * 32 + i]);
    float s = 0.f;
    for (int i = 0; i < V_EXO; ++i) { float e = __expf(sc[tid * 32 + i] - m); att[tid * 32 + i] = e; s += e; }
    float inv = 1.f / s;
    for (int i = 0; i < V_EXO; ++i) att[tid * 32 + i] *= inv;
  }
  __syncthreads();
  for (int d = tid; d < 512; d += 256) {
    int hh = d >> 6;
    float a = 0.f;
    for (int i = 0; i < V_EXO; ++i) a += att[hh * 32 + i] * vbuf[((size_t)(b * V_EXO + i)) * 512 + d];
    ov[d] = a;
  }
  __syncthreads();
  for (int o = tid; o < 512; o += 256) {
    float a = out_b[o];
    for (int d = 0; d < 512; ++d) a += ov[d] * out_w[(size_t)o * 512 + d];
    rbuf[o] = a + g_endo[o];
  }
  __syncthreads();
  float s = 0.f, s2 = 0.f;
  for (int o = tid; o < 512; o += 256) { float x = rbuf[o]; s += x; s2 += x * x; }
  for (int off = 16; off; off >>= 1) { s += __shfl_xor(s, off, 32); s2 += __shfl_xor(s2, off, 32); }
  if ((tid & 31) == 0) { rs[tid >> 5] = s; rs2[tid >> 5] = s2; }
  __syncthreads();
  if (tid == 0) {
    float ts = 0.f, t2 = 0.f;
    for (int w = 0; w < 8; ++w) { ts += rs[w]; t2 += rs2[w]; }
    float mean = ts * (1.f / 512.f);
    float var = t2 * (1.f / 512.f) - mean * mean;
    mv[0] = mean; mv[1] = rsqrtf(var + 1e-5f);
  }
  __syncthreads();
  float mean = mv[0], rstd = mv[1];
  for (int o = tid; o < 512; o += 256)
    tokens[((size_t)(b * NTOK)) * 512 + o] = (rbuf[o] - mean) * rstd * ln_g[o] + ln_b[o];
}

// ---------------------------------------------------------------------------
// Encoder depthwise conv (K=2) over token axis + silu
// ---------------------------------------------------------------------------
__global__ __launch_bounds__(256) void enc_conv_kernel(
    const float* __restrict__ xi_z, const float* __restrict__ conv_w,
    const float* __restrict__ conv_b, float* __restrict__ xc)
{
  size_t idx = (size_t)blockIdx.x * 256 + threadIdx.x;
  if (idx >= (size_t)NROWS * D_MODEL) return;
  int d = idx & 511;
  size_t row = idx >> 9;
  int l = (int)(row % NTOK);
  float cur  = xi_z[row * 1024 + d];
  float prev = (l == 0) ? 0.f : xi_z[(row - 1) * 1024 + d];
  float pre = conv_w[d] * prev + conv_w[512 + d] * cur + conv_b[d];
  xc[idx] = siluf(pre);
}

// ---------------------------------------------------------------------------
// Encoder selective scan: grid (32 batches x 8 channel-slices of 64).
// B/C/dt/xc for all L=49 staged in LDS; h[64], A[64] in registers.
// Writes gated output u = (y + xc*D) * silu(z).
// ---------------------------------------------------------------------------
__global__ __launch_bounds__(256) void enc_scan_kernel(
    const float* __restrict__ dbc, const float* __restrict__ dtb,
    const float* __restrict__ xc,  const float* __restrict__ xi_z,
    const float* __restrict__ A_log, const float* __restrict__ Dv,
    float* __restrict__ ubuf)
{
  extern __shared__ float sm[];
  float* sB  = sm;                     // 49*256
  float* sC  = sB + NTOK * D_STATE;    // 49*256
  float* sdt = sC + NTOK * D_STATE;    // 49*64
  float* sxc = sdt + NTOK * 64;        // 49*64

  const int b = blockIdx.x;
  const int ds0 = (int)blockIdx.y * 64;
  const int tid = threadIdx.x;

  for (int i = tid; i < NTOK * D_STATE; i += 256) {
    int l = i >> 8, s = i & 255;
    size_t rb = (size_t)(b * NTOK + l) * DBC_M;
    sB[i] = dbc[rb + DTRK + s];
    sC[i] = dbc[rb + DTRK + D_STATE + s];
  }
  for (int i = tid; i < NTOK * 64; i += 256) {
    int l = i >> 6, d = i & 63;
    size_t r = (size_t)(b * NTOK + l) * D_MODEL;
    sdt[i] = dtb[r + ds0 + d];
    sxc[i] = xc[r + ds0 + d];
  }
  const int dl = tid >> 2, squad = tid & 3, s0 = squad * 64;
  float h[64], Areg[64];
  const float* Arow = A_log + (size_t)(ds0 + dl) * D_STATE + s0;
#pragma unroll
  for (int j = 0; j < 64; ++j) { h[j] = 0.f; Areg[j] = -__expf(Arow[j]); }
  const float Dval = Dv[ds0 + dl];
  __syncthreads();

  for (int l = 0; l < NTOK; ++l) {
    const float dtv = sdt[l * 64 + dl];
    const float xv  = sxc[l * 64 + dl];
    const float dtx = dtv * xv;
    float acc = 0.f;
    const float* bl = sB + l * D_STATE + s0;
    const float* cl = sC + l * D_STATE + s0;
#pragma unroll
    for (int j = 0; j < 64; ++j) {
      float dA = __expf(dtv * Areg[j]);
      h[j] = dA * h[j] + dtx * bl[j];
      acc += h[j] * cl[j];
    }
    acc += __shfl_xor(acc, 1, 32);
    acc += __shfl_xor(acc, 2, 32);
    if (squad == 0) {
      size_t r = (size_t)(b * NTOK + l);
      float y = acc + xv * Dval;
      float z = xi_z[r * 1024 + D_MODEL + ds0 + dl];
      ubuf[r * D_MODEL + ds0 + dl] = y * siluf(z);
    }
  }
}

// ---------------------------------------------------------------------------
// Residual add + LayerNorm (in place on x), D = 512, one row per block
// ---------------------------------------------------------------------------
__global__ __launch_bounds__(256) void add_ln_kernel(
    float* __restrict__ x, const float* __restrict__ y,
    const float* __restrict__ g, const float* __restrict__ b)
{
  __shared__ float rs[8], rs2[8], mv[2];
  const int row = blockIdx.x, tid = threadIdx.x;
  const size_t base = (size_t)row * 512;
  float v0 = x[base + tid]       + y[base + tid];
  float v1 = x[base + 256 + tid] + y[base + 256 + tid];
  float s = v0 + v1, s2 = v0 * v0 + v1 * v1;
  for (int off = 16; off; off >>= 1) { s += __shfl_xor(s, off, 32); s2 += __shfl_xor(s2, off, 32); }
  if ((tid & 31) == 0) { rs[tid >> 5] = s; rs2[tid >> 5] = s2; }
  __syncthreads();
  if (tid == 0) {
    float ts = 0.f, t2 = 0.f;
    for (int w = 0; w < 8; ++w) { ts += rs[w]; t2 += rs2[w]; }
    float mean = ts * (1.f / 512.f);
    float var = t2 * (1.f / 512.f) - mean * mean;
    mv[0] = mean; mv[1] = rsqrtf(var + 1e-5f);
  }
  __syncthreads();
  float mean = mv[0], rstd = mv[1];
  x[base + tid]       = (v0 - mean) * rstd * g[tid]       + b[tid];
  x[base + 256 + tid] = (v1 - mean) * rstd * g[256 + tid] + b[256 + tid];
}

// ---------------------------------------------------------------------------
// Heads: y[b,t] = mean_p(tok[b,1+p]) @ ehw + ehb + tok[b,0] @ xhw + xhb,
// broadcast over vars. out: (32, 7, 32)
// ---------------------------------------------------------------------------
__global__ __launch_bounds__(256) void heads_kernel(
    const float* __restrict__ tokens,
    const float* __restrict__ ehw, const float* __restrict__ ehb,
    const float* __restrict__ xhw, const float* __restrict__ xhb,
    float* __restrict__ out)
{
  __shared__ float part[256 * 8];
  __shared__ float yv[8];
  const int b = blockIdx.x, tid = threadIdx.x;
  float p[7];
#pragma unroll
  for (int t = 0; t < 7; ++t) p[t] = 0.f;
  for (int d = tid; d < 512; d += 256) {
    float ef = 0.f;
    for (int pp = 0; pp < P_PATCH; ++pp)
      ef += tokens[((size_t)(b * NTOK + 1 + pp)) * 512 + d];
    ef *= (1.f / (float)P_PATCH);
    float gf = tokens[((size_t)(b * NTOK)) * 512 + d];
#pragma unroll
    for (int t = 0; t < 7; ++t) p[t] += ef * ehw[d * 7 + t] + gf * xhw[d * 7 + t];
  }
#pragma unroll
  for (int t = 0; t < 7; ++t) part[tid * 8 + t] = p[t];
  __syncthreads();
  for (int stride = 128; stride; stride >>= 1) {
    if (tid < stride) {
#pragma unroll
      for (int t = 0; t < 7; ++t) part[tid * 8 + t] += part[(tid + stride) * 8 + t];
    }
    __syncthreads();
  }
  if (tid < 7) yv[tid] = part[tid] + ehb[tid] + xhb[tid];
  __syncthreads();
  for (int i = tid; i < PRED_LEN * N_VARS; i += 256) {
    int t = i / N_VARS, v = i % N_VARS;
    out[((size_t)b * PRED_LEN + t) * N_VARS + v] = yv[t];
  }
}

// ---------------------------------------------------------------------------
// Host side
// ---------------------------------------------------------------------------
static void launch_gemm(hipStream_t st, const float* A, int lda,
                        const float* B, int ldb, int transB, const float* bias,
                        float* C, int ldc, int M, int N, int K, int act)
{
  dim3 g((M + 63) / 64, (N + 63) / 64);
  gemm_wmma_f32<<<g, 256, 0, st>>>(A, lda, B, ldb, transB, bias, C, ldc, M, N, K, act);
}

// Input flattening (jax pytree order: sorted dict keys, lists in sequence):
// 0 x_enc
// enc[i] at 1+i*17: ffn_b1,ffn_b2,ffn_w1,ffn_w2,ln1_b,ln1_g,ln2_b,ln2_g,
//                   A_log,D,conv_b,conv_w,dt_b,dt_w,in_w,out_w,x_w
// 52 endo.g, 53 endo.pos, 54 endo.proj_b, 55 endo.proj_w
// 56 endo_head_b, 57 endo_head_w
// 58 exo.in_b, 59 exo.in_w,
// 60..68 exo.mamba: A_log,D,conv_b,conv_w,dt_b,dt_w,in_w,out_w,x_w
// 69 exo.out_b, 70 exo.out_w, 71 exo_head_b, 72 exo_head_w
// 73 fuse.in_b, 74 fuse.in_w, 75 fuse.ln_b, 76 fuse.ln_g, 77 fuse.out_b, 78 fuse.out_w
extern "C" void kernel_launch(void* const* d_in, const int* in_sizes, int n_in,
                              void* d_out, int out_size, void* d_ws, size_t ws_size,
                              hipStream_t stream)
{
  (void)in_sizes; (void)n_in; (void)out_size; (void)ws_size;
  const float* x_enc = (const float*)d_in[0];
  auto P = [&](int i) { return (const float*)d_in[i]; };

  float* ws = (float*)d_ws;
  size_t o = 0;
  float* tokens  = ws + o; o += (size_t)NROWS * D_MODEL;   // 1568x512
  float* xi_z    = ws + o; o += (size_t)NROWS * 1024;      // 1568x1024
  float* xc      = ws + o; o += (size_t)NROWS * D_MODEL;
  float* dbc     = ws + o; o += (size_t)NROWS * DBC_M;     // 1568x544
  float* dtb     = ws + o; o += (size_t)NROWS * D_MODEL;
  float* ubuf    = ws + o; o += (size_t)NROWS * D_MODEL;
  float* ffn     = ws + o; o += (size_t)NROWS * D_FF;      // 1568x2048
  float* ym      = ws + o; o += (size_t)NROWS * D_MODEL;
  float* exo_tok = ws + o; o += (size_t)NSEQ * D_MODEL;    // 992x512
  float* kbuf    = ws + o; o += (size_t)NSEQ * D_MODEL;
  float* vbuf    = ws + o; o += (size_t)NSEQ * D_MODEL;
  float* u2c2    = ws + o; o += 256;
  float* W2      = ws + o; o += 64 * 512;
  float* qv      = ws + o; o += 512;

  // 1) precompute fused small matrices / vectors
  prep_kernel<<<1, 256, 0, stream>>>(P(59), P(58), P(66), P(67), P(70),
                                     P(52), P(74), P(73), u2c2, W2, qv);

  // 2) endo patch tokens (rows b*49 + 1 + p)
  endo_tokens_kernel<<<((size_t)BATCH * P_PATCH * D_MODEL + 255) / 256, 256, 0, stream>>>(
      x_enc, P(55), P(54), P(53), tokens);

  // 3) exo Mamba (992 sequences) -> exo_tok
  {
    size_t lds = (size_t)(SEQ_LEN + LCH * 64 + LCH * 4 + LCH * 64 + LCH * 128 + LCH * 128) * 4;
    exo_mamba_kernel<<<NSEQ, 256, lds, stream>>>(
        x_enc, u2c2, P(63), P(62), P(68), P(65), P(64), P(60), P(61), W2, P(69), exo_tok);
  }

  // 4) k/v projections (WMMA, B transposed: W (512,512) row-major, x @ W.T)
  launch_gemm(stream, exo_tok, 512, P(74) + (size_t)512 * 512, 512, 1, P(73) + 512,
              kbuf, 512, NSEQ, 512, 512, 0);
  launch_gemm(stream, exo_tok, 512, P(74) + (size_t)1024 * 512, 512, 1, P(73) + 1024,
              vbuf, 512, NSEQ, 512, 512, 0);

  // 5) fused cross-attention + out proj + residual + LN -> tokens row b*49
  fuse_attn_kernel<<<BATCH, 256, 0, stream>>>(qv, kbuf, vbuf, P(78), P(77), P(52),
                                              P(76), P(75), tokens);

  // 6) encoder layers
  for (int L = 0; L < 3; ++L) {
    int base = 1 + L * 17;
    const float *ffn_b1 = P(base + 0), *ffn_b2 = P(base + 1);
    const float *ffn_w1 = P(base + 2), *ffn_w2 = P(base + 3);
    const float *ln1_b = P(base + 4), *ln1_g = P(base + 5);
    const float *ln2_b = P(base + 6), *ln2_g = P(base + 7);
    const float *A_log = P(base + 8), *Dv = P(base + 9);
    const float *conv_b = P(base + 10), *conv_w = P(base + 11);
    const float *dt_b = P(base + 12), *dt_w = P(base + 13);
    const float *in_w = P(base + 14), *out_w = P(base + 15), *x_w = P(base + 16);

    launch_gemm(stream, tokens, 512, in_w, 1024, 0, nullptr, xi_z, 1024, NROWS, 1024, 512, 0);
    enc_conv_kernel<<<((size_t)NROWS * D_MODEL + 255) / 256, 256, 0, stream>>>(
        xi_z, conv_w, conv_b, xc);
    launch_gemm(stream, xc, 512, x_w, DBC_M, 0, nullptr, dbc, DBC_M, NROWS, DBC_M, 512, 0);
    launch_gemm(stream, dbc, DBC_M, dt_w, 512, 0, dt_b, dtb, 512, NROWS, 512, DTRK, 2); // softplus
    {
      size_t lds = (size_t)(NTOK * D_STATE * 2 + NTOK * 64 * 2) * 4;
      enc_scan_kernel<<<dim3(BATCH, 8), 256, lds, stream>>>(dbc, dtb, xc, xi_z, A_log, Dv, ubuf);
    }
    launch_gemm(stream, ubuf, 512, out_w, 512, 0, nullptr, ym, 512, NROWS, 512, 512, 0);
    add_ln_kernel<<<NROWS, 256, 0, stream>>>(tokens, ym, ln1_g, ln1_b);
    launch_gemm(stream, tokens, 512, ffn_w1, D_FF, 0, ffn_b1, ffn, D_FF, NROWS, D_FF, 512, 1); // relu
    launch_gemm(stream, ffn, D_FF, ffn_w2, 512, 0, ffn_b2, ym, 512, NROWS, 512, D_FF, 0);
    add_ln_kernel<<<NROWS, 256, 0, stream>>>(tokens, ym, ln2_g, ln2_b);
  }

  // 7) heads -> (32, 7, 32)
  heads_kernel<<<BATCH, 256, 0, stream>>>(tokens, P(57), P(56), P(72), P(71), (float*)d_out);
}